// COLT_56873956933770
// MI455X (gfx1250) — compile-verified
//
#include <hip/hip_runtime.h>

// Problem constants (match the reference)
#define COLT_NQ 100000
#define COLT_NT 20000
#define COLT_NS 10000
#define COLT_D  384

typedef __attribute__((ext_vector_type(2))) float v2f;
typedef __attribute__((ext_vector_type(8))) float v8f;

// ---------------------------------------------------------------------------
// SpMM (COO, scatter form): out[r,:] += v * x[c,:]
// 4 edges per 384-thread block; each lane handles one float4 quad of the row.
// Each wave32 lies entirely inside one edge -> the dropout (v==0) skip is
// wave-uniform. Scatter via hardware fp32 global atomics (non-returning).
// ---------------------------------------------------------------------------
__global__ __launch_bounds__(384)
void colt_spmm_kernel(const int* __restrict__ rows, const int* __restrict__ cols,
                      const float* __restrict__ vals, const float* __restrict__ x,
                      float* __restrict__ out, int nnz)
{
    const int QPR = COLT_D / 4;               // 96 quads per row
    const int lid = threadIdx.x;              // 0..383
    const int e   = blockIdx.x * 4 + lid / QPR;
    const int q   = lid % QPR;
    if (e >= nnz) return;
    const float v = vals[e];
    if (v == 0.0f) return;                    // edge-dropout: wave-uniform skip
    const size_t c = (size_t)cols[e];
    const size_t r = (size_t)rows[e];
    const float4 xv = *(const float4*)(x + c * COLT_D + (size_t)q * 4);
    float* op = out + r * COLT_D + (size_t)q * 4;
    unsafeAtomicAdd(op + 0, v * xv.x);
    unsafeAtomicAdd(op + 1, v * xv.y);
    unsafeAtomicAdd(op + 2, v * xv.z);
    unsafeAtomicAdd(op + 3, v * xv.w);
}

// ---------------------------------------------------------------------------
// Per-row scale + L2-normalize + accumulate, 16 rows per wave.
// f = tmp_row * invdiv;  feats_row = f;  acc_row += f / max(||f||, 1e-12)
//
// Row norms via v_wmma_f32_16x16x4_f32: for the f32 16x16x4 shape the A
// (16x4) and B (4x16) VGPR layouts coincide under transpose, so feeding the
// same registers as A and B accumulates the Gram matrix F*F^T; its diagonal
// is the squared row norms. Tile is staged in LDS for the second pass.
// ---------------------------------------------------------------------------
#define COLT_NW 2   // waves (16-row tiles) per block -> 48 KB LDS

__global__ __launch_bounds__(32 * COLT_NW)
void colt_norm_acc_kernel(const float* __restrict__ tmp, float* __restrict__ feats,
                          float* __restrict__ acc, float invdiv, int ntiles)
{
    __shared__ __attribute__((aligned(16))) float tileLds[COLT_NW][16 * COLT_D];
    __shared__ float norms[COLT_NW][16];

    const int lane = threadIdx.x & 31;
    const int wid  = threadIdx.x >> 5;
    const int tile = blockIdx.x * COLT_NW + wid;
    const bool active = (tile < ntiles);       // wave-uniform guard

    if (active) {
        const int r0  = tile * 16;
        const int rit = lane & 15;             // row within tile (M)
        const int kh  = (lane >> 4) << 1;      // K half: lanes<16 -> {0,1}, else {2,3}
        const float* rowp = tmp + (size_t)(r0 + rit) * COLT_D + kh;
        float* lrow = &tileLds[wid][rit * COLT_D + kh];
        v8f cacc = {};
        #pragma unroll 8
        for (int kb = 0; kb < COLT_D; kb += 4) {
            v2f a = *(const v2f*)(rowp + kb);
            a = a * invdiv;
            *(v2f*)(lrow + kb) = a;
            // C += A * A^T  (same regs serve as A and B for the f32 x4 shape)
            cacc = __builtin_amdgcn_wmma_f32_16x16x4_f32(
                false, a, false, a, (short)0, cacc, false, false);
        }
        // Diagonal of C: row j (j<8) at vgpr j / lane j; row 8+j at vgpr j / lane 24+j
        float n2 = 0.0f;
        #pragma unroll
        for (int j = 0; j < 8; ++j) {
            n2 = (lane == j)      ? cacc[j] : n2;
            n2 = (lane == 24 + j) ? cacc[j] : n2;
        }
        if (lane < 8)        norms[wid][lane]      = n2;
        else if (lane >= 24) norms[wid][lane - 16] = n2;
    }
    __syncthreads();
    if (active) {
        const int r0 = tile * 16;
        for (int idx = lane; idx < 16 * (COLT_D / 4); idx += 32) {
            const int row = idx / (COLT_D / 4);
            const int q   = idx % (COLT_D / 4);
            const float rn = 1.0f / fmaxf(sqrtf(norms[wid][row]), 1e-12f);
            const float4 f = *(const float4*)&tileLds[wid][row * COLT_D + q * 4];
            const size_t go = (size_t)(r0 + row) * COLT_D + (size_t)q * 4;
            float4 av = *(float4*)(acc + go);
            av.x += f.x * rn; av.y += f.y * rn;
            av.z += f.z * rn; av.w += f.w * rn;
            *(float4*)(acc + go) = av;            // acc rows unique per tile: no atomics
            *(float4*)(feats + go) = f;           // un-normalized feats for next layer
        }
    }
}

// ---------------------------------------------------------------------------
extern "C" void kernel_launch(void* const* d_in, const int* in_sizes, int n_in,
                              void* d_out, int out_size, void* d_ws, size_t ws_size,
                              hipStream_t stream)
{
    const float* qf        = (const float*)d_in[0];
    const float* tf        = (const float*)d_in[1];
    const float* tl_vals   = (const float*)d_in[2];
    const float* sl_vals   = (const float*)d_in[3];
    const float* agg_vals  = (const float*)d_in[4];
    const float* aggo_vals = (const float*)d_in[5];
    const int* tl_rows  = (const int*)d_in[6];
    const int* tl_cols  = (const int*)d_in[7];
    const int* sl_rows  = (const int*)d_in[8];
    const int* sl_cols  = (const int*)d_in[9];
    const int* agg_rows = (const int*)d_in[10];
    const int* agg_cols = (const int*)d_in[11];
    const int tl_nnz  = in_sizes[6];
    const int sl_nnz  = in_sizes[8];
    const int agg_nnz = in_sizes[10];

    const size_t eQ = (size_t)COLT_NQ * COLT_D;
    const size_t eT = (size_t)COLT_NT * COLT_D;
    const size_t eS = (size_t)COLT_NS * COLT_D;

    float* out = (float*)d_out;
    float* TLq = out;                    // acc for TL propagation ([q; t] contiguous)
    float* TLt = out + eQ;
    float* TLs = out + eQ + eT;
    float* SLq = out + eQ + eT + eS;     // acc for SL propagation ([q; s] contiguous)

    float* fbuf = (float*)d_ws;          // layer features, (NQ+NT)*D floats
    float* tbuf = fbuf + (eQ + eT);      // spmm target,    (NQ+NT)*D floats
    float* sbuf = tbuf + (eQ + eT);      // scenes_feature, NS*D floats

    auto spmm = [&](const int* r, const int* c, const float* v, const float* x,
                    float* o, int nnz) {
        colt_spmm_kernel<<<dim3((nnz + 3) / 4), 384, 0, stream>>>(r, c, v, x, o, nnz);
    };
    auto normacc = [&](const float* t, float* f, float* a, float inv, int nrows) {
        const int ntiles = nrows / 16;   // 120000 and 110000 are multiples of 16
        colt_norm_acc_kernel<<<dim3((ntiles + COLT_NW - 1) / COLT_NW),
                               32 * COLT_NW, 0, stream>>>(t, f, a, inv, ntiles);
    };

    // 1) scenes_feature = agg_ori @ tools_feature
    hipMemsetAsync(sbuf, 0, eS * sizeof(float), stream);
    spmm(agg_rows, agg_cols, aggo_vals, tf, sbuf, agg_nnz);

    // 2) TL propagation over [queries; tools]
    hipMemcpyAsync(fbuf,      qf, eQ * sizeof(float), hipMemcpyDeviceToDevice, stream);
    hipMemcpyAsync(fbuf + eQ, tf, eT * sizeof(float), hipMemcpyDeviceToDevice, stream);
    hipMemcpyAsync(TLq, fbuf, (eQ + eT) * sizeof(float), hipMemcpyDeviceToDevice, stream);
    for (int i = 0; i < 2; ++i) {
        hipMemsetAsync(tbuf, 0, (eQ + eT) * sizeof(float), stream);
        spmm(tl_rows, tl_cols, tl_vals, fbuf, tbuf, tl_nnz);
        normacc(tbuf, fbuf, TLq, 1.0f / (float)(i + 2), COLT_NQ + COLT_NT);
    }

    // 3) TL_s = agg(ED-dropped) @ TL_t
    hipMemsetAsync(TLs, 0, eS * sizeof(float), stream);
    spmm(agg_rows, agg_cols, agg_vals, TLt, TLs, agg_nnz);

    // 4) SL propagation over [queries; scenes]
    hipMemcpyAsync(fbuf,      qf,   eQ * sizeof(float), hipMemcpyDeviceToDevice, stream);
    hipMemcpyAsync(fbuf + eQ, sbuf, eS * sizeof(float), hipMemcpyDeviceToDevice, stream);
    hipMemcpyAsync(SLq, fbuf, (eQ + eS) * sizeof(float), hipMemcpyDeviceToDevice, stream);
    for (int i = 0; i < 2; ++i) {
        hipMemsetAsync(tbuf, 0, (eQ + eS) * sizeof(float), stream);
        spmm(sl_rows, sl_cols, sl_vals, fbuf, tbuf, sl_nnz);
        normacc(tbuf, fbuf, SLq, 1.0f / (float)(i + 2), COLT_NQ + COLT_NS);
    }

    (void)n_in; (void)out_size; (void)ws_size;
}